// TransformerCell_74758200754286
// MI455X (gfx1250) — compile-verified
//
#include <hip/hip_runtime.h>

// ---------------------------------------------------------------------------
// TransformerCell for MI455X (gfx1250, wave32, WMMA).
//
// Incremental-attention rewrite: maintain K = Z*W2^T, P = K*W1^T, V = Z*W3^T
// and update one row per step (exact — unwritten Z rows are zero, so their
// P/V rows are zero; their softmax contribution is (257-t)*exp(-m), folded in
// analytically). Per step: 3 update matvecs + t dots + softmax + g-reduce +
// 4 MLP matvecs. Everything is f32 and L2-resident (~24 MB).
//
// Single persistent kernel (8 blocks x 256 thr = 64 waves) with an atomic
// sense-reversing grid barrier; all GEMVs use V_WMMA_F32_16X16X4_F32 with the
// vector in column 0 of B (full-f32 precision, 64 weights per WMMA in A).
//
// Workspace use: ~1.6 MB of d_ws.
// ---------------------------------------------------------------------------

#define N_T   256
#define N_LAT 1024
#define N_INN 512
#define N_G   512
#define N_MU  64
#define N_NEU 1024
#define NROWS (N_T + 1)          // 257

#define NBLOCKS 8
#define TPB     256
#define NWAVES  ((NBLOCKS * TPB) / 32)   // 64 wave32s

typedef float v2f __attribute__((ext_vector_type(2)));
typedef float v8f __attribute__((ext_vector_type(8)));

struct Params {
  const float *mu, *z0;
  const float *W1, *W2, *W3;
  const float *p_in_w, *p_in_b, *p_h0_w, *p_h0_b, *p_h1_w, *p_h1_b, *p_out_w, *p_out_b;
  const float *m_in_w, *m_in_b, *m_h0_w, *m_h0_b, *m_h1_w, *m_h1_b, *m_out_w, *m_out_b;
  float *Z;        // d_out: [257][1024]
  float *P;        // ws: [257][1024]
  float *V;        // ws: [257][512]
  float *Kt;       // ws: [2][512]
  float *g;        // ws: [512]
  float *h0;       // ws: [1024]
  float *h1;       // ws: [1024]
  float *logits;   // ws: [257]
  float *a;        // ws: [257]
  unsigned *bar;   // ws: [2] {count, generation}
};

// ---------------- device-wide sense-reversing barrier ----------------------
__device__ inline void grid_sync(unsigned *bar) {
  __threadfence();           // release this thread's global writes (agent scope)
  __syncthreads();
  if (threadIdx.x == 0) {
    unsigned *cnt = bar, *gen = bar + 1;
    unsigned g = __hip_atomic_load(gen, __ATOMIC_RELAXED, __HIP_MEMORY_SCOPE_AGENT);
    unsigned arrived =
        __hip_atomic_fetch_add(cnt, 1u, __ATOMIC_ACQ_REL, __HIP_MEMORY_SCOPE_AGENT) + 1u;
    if (arrived == gridDim.x) {
      __hip_atomic_store(cnt, 0u, __ATOMIC_RELAXED, __HIP_MEMORY_SCOPE_AGENT);
      __hip_atomic_fetch_add(gen, 1u, __ATOMIC_RELEASE, __HIP_MEMORY_SCOPE_AGENT);
    } else {
      while (__hip_atomic_load(gen, __ATOMIC_ACQUIRE, __HIP_MEMORY_SCOPE_AGENT) == g) {
        __builtin_amdgcn_s_sleep(2);
      }
    }
  }
  __syncthreads();
}

// ---------------- grid-cooperative GEMV via V_WMMA_F32_16X16X4_F32 ---------
// y[0..outF) = act(W @ x + bias) (+ residual). W row-major [outF x inF].
// Each wave owns 16-row output tiles. A tile = A(16x4 f32) per k-step:
//   lane l (l<16):   A = { W[r0+l][k0],   W[r0+l][k0+1] }
//   lane 16+l:       A = { W[r0+l][k0+2], W[r0+l][k0+3] }
// B(4x16) holds x in column 0 only (lane 0: x[k0..k0+1], lane 16: x[k0+2..3]).
// Column 0 of D: lane 0 -> rows r0..r0+7, lane 16 -> rows r0+8..r0+15.
// Must be called from wave-uniform control flow with full EXEC (WMMA req).
__device__ inline void matvec_wmma(const float *__restrict__ W,
                                   const float *__restrict__ x,
                                   const float *__restrict__ bias,
                                   const float *__restrict__ residual,
                                   float *__restrict__ y,
                                   int outF, int inF, bool leaky,
                                   int gwave, int lane) {
  const int m      = lane & 15;            // output row within 16-row tile
  const int khalf  = (lane >> 4) << 1;     // 0 or 2 (K split across lane halves)
  const float bm   = (m == 0) ? 1.0f : 0.0f;  // column-0 mask for B
  const int njobs  = outF >> 4;
  for (int job = gwave; job < njobs; job += NWAVES) {
    const int r0 = job << 4;
    const float *Wr = W + (size_t)(r0 + m) * inF + khalf;
    v8f c = {};
    for (int k0 = 0; k0 < inF; k0 += 4) {
      v2f a; a.x = Wr[k0];                 a.y = Wr[k0 + 1];
      v2f b; b.x = x[k0 + khalf] * bm;     b.y = x[k0 + khalf + 1] * bm;
      c = __builtin_amdgcn_wmma_f32_16x16x4_f32(
              /*neg_a=*/false, a, /*neg_b=*/false, b,
              /*c_mod=*/(short)0, c, /*reuse_a=*/false, /*reuse_b=*/false);
    }
    if (m == 0) {                          // lanes 0 and 16 hold column 0 of D
      const int rb = r0 + ((lane >> 4) << 3);
#pragma unroll
      for (int j = 0; j < 8; ++j) {
        float v = c[j];
        if (bias)     v += bias[rb + j];
        if (leaky)    v = (v > 0.0f) ? v : 0.01f * v;   // jax leaky_relu slope
        if (residual) v += residual[rb + j];
        y[rb + j] = v;
      }
    }
  }
}

// ---------------- wave32 dot product ---------------------------------------
__device__ inline float wave_dot(const float *__restrict__ a,
                                 const float *__restrict__ b, int n, int lane) {
  float s = 0.0f;
  for (int i = lane; i < n; i += 32) s += a[i] * b[i];
#pragma unroll
  for (int off = 16; off >= 1; off >>= 1) s += __shfl_down(s, off, 32);
  return s;  // valid in lane 0
}

// ---------------- kernels ---------------------------------------------------
__global__ void init_bar_kernel(unsigned *bar) { bar[0] = 0u; bar[1] = 0u; }

__global__ __launch_bounds__(TPB) void cell_kernel(Params p) {
  const int lane    = threadIdx.x & 31;
  const int gtid    = blockIdx.x * blockDim.x + threadIdx.x;
  const int gwave   = gtid >> 5;
  const int nthread = gridDim.x * blockDim.x;
  const float inv_sqrt = 0.03125f;  // 1/sqrt(1024), exact

  // ---- Phase 0: z_minus = MLP_p(mu); Z[0]=z_minus, Z[1]=z_0 ----
  matvec_wmma(p.p_in_w, p.mu, p.p_in_b, nullptr, p.h0, N_NEU, N_MU, true, gwave, lane);
  grid_sync(p.bar);
  matvec_wmma(p.p_h0_w, p.h0, p.p_h0_b, nullptr, p.h1, N_NEU, N_NEU, true, gwave, lane);
  grid_sync(p.bar);
  matvec_wmma(p.p_h1_w, p.h1, p.p_h1_b, nullptr, p.h0, N_NEU, N_NEU, true, gwave, lane);
  grid_sync(p.bar);
  matvec_wmma(p.p_out_w, p.h0, p.p_out_b, nullptr, p.Z, N_LAT, N_NEU, false, gwave, lane);
  for (int i = gtid; i < N_LAT; i += nthread) p.Z[N_LAT + i] = p.z0[i];
  grid_sync(p.bar);

  // ---- Phase 1: K/V for rows 0,1; then P rows 0,1 ----
  matvec_wmma(p.W2, p.Z,         nullptr, nullptr, p.Kt,          N_INN, N_LAT, false, gwave, lane);
  matvec_wmma(p.W2, p.Z + N_LAT, nullptr, nullptr, p.Kt + N_INN,  N_INN, N_LAT, false, gwave, lane);
  matvec_wmma(p.W3, p.Z,         nullptr, nullptr, p.V,           N_G,   N_LAT, false, gwave, lane);
  matvec_wmma(p.W3, p.Z + N_LAT, nullptr, nullptr, p.V + N_G,     N_G,   N_LAT, false, gwave, lane);
  grid_sync(p.bar);
  matvec_wmma(p.W1, p.Kt,         nullptr, nullptr, p.P,          N_LAT, N_INN, false, gwave, lane);
  matvec_wmma(p.W1, p.Kt + N_INN, nullptr, nullptr, p.P + N_LAT,  N_LAT, N_INN, false, gwave, lane);
  grid_sync(p.bar);

  // ---- Phase 2: recurrence, t = 2..256 ----
  for (int t = 2; t <= N_T; ++t) {
    const float *z_prev = p.Z + (size_t)(t - 1) * N_LAT;

    // logits[i] = dot(P[i], z_prev) / 32   (rows i >= t are exactly 0)
    for (int i = gwave; i < t; i += NWAVES) {
      float d = wave_dot(p.P + (size_t)i * N_LAT, z_prev, N_LAT, lane);
      if (lane == 0) p.logits[i] = d * inv_sqrt;
    }
    grid_sync(p.bar);

    // softmax over 257 entries; (257 - t) implicit zero logits handled exactly
    if (gwave == 0) {
      float mx = 0.0f;  // max includes the implicit zeros
      for (int i = lane; i < t; i += 32) mx = fmaxf(mx, p.logits[i]);
#pragma unroll
      for (int off = 16; off >= 1; off >>= 1) mx = fmaxf(mx, __shfl_xor(mx, off, 32));
      float s = 0.0f;
      for (int i = lane; i < t; i += 32) s += expf(p.logits[i] - mx);
#pragma unroll
      for (int off = 16; off >= 1; off >>= 1) s += __shfl_xor(s, off, 32);
      s += (float)(NROWS - t) * expf(-mx);
      const float inv = 1.0f / s;
      for (int i = lane; i < t; i += 32) p.a[i] = expf(p.logits[i] - mx) * inv;
    }
    grid_sync(p.bar);

    // g[j] = sum_{i<t} a[i] * V[i][j]   (rows >= t are zero: skipped)
    for (int j = gtid; j < N_G; j += nthread) {
      float acc = 0.0f;
      for (int i = 0; i < t; ++i) acc += p.a[i] * p.V[(size_t)i * N_G + j];
      p.g[j] = acc;
    }
    grid_sync(p.bar);

    // z_new = z_prev + MLP_m(g); write Z[t]
    matvec_wmma(p.m_in_w, p.g,  p.m_in_b, nullptr, p.h0, N_NEU, N_G,   true, gwave, lane);
    grid_sync(p.bar);
    matvec_wmma(p.m_h0_w, p.h0, p.m_h0_b, nullptr, p.h1, N_NEU, N_NEU, true, gwave, lane);
    grid_sync(p.bar);
    matvec_wmma(p.m_h1_w, p.h1, p.m_h1_b, nullptr, p.h0, N_NEU, N_NEU, true, gwave, lane);
    grid_sync(p.bar);
    matvec_wmma(p.m_out_w, p.h0, p.m_out_b, z_prev,
                p.Z + (size_t)t * N_LAT, N_LAT, N_NEU, false, gwave, lane);
    grid_sync(p.bar);

    // Extend caches with row t (needed for step t+1; skip after last step)
    if (t < N_T) {
      const float *z_new = p.Z + (size_t)t * N_LAT;
      matvec_wmma(p.W2, z_new, nullptr, nullptr, p.Kt,                    N_INN, N_LAT, false, gwave, lane);
      matvec_wmma(p.W3, z_new, nullptr, nullptr, p.V + (size_t)t * N_G,   N_G,   N_LAT, false, gwave, lane);
      grid_sync(p.bar);
      matvec_wmma(p.W1, p.Kt, nullptr, nullptr, p.P + (size_t)t * N_LAT,  N_LAT, N_INN, false, gwave, lane);
      grid_sync(p.bar);
    }
  }
}

// ---------------- host entry ------------------------------------------------
extern "C" void kernel_launch(void *const *d_in, const int *in_sizes, int n_in,
                              void *d_out, int out_size, void *d_ws, size_t ws_size,
                              hipStream_t stream) {
  (void)in_sizes; (void)n_in; (void)out_size; (void)ws_size;
  const float *const *in = (const float *const *)d_in;
  Params p;
  p.mu = in[0];  p.z0 = in[1];
  p.W1 = in[2];  p.W2 = in[3];  p.W3 = in[4];
  p.p_in_w = in[5];  p.p_in_b = in[6];  p.p_h0_w = in[7];  p.p_h0_b = in[8];
  p.p_h1_w = in[9];  p.p_h1_b = in[10]; p.p_out_w = in[11]; p.p_out_b = in[12];
  p.m_in_w = in[13]; p.m_in_b = in[14]; p.m_h0_w = in[15]; p.m_h0_b = in[16];
  p.m_h1_w = in[17]; p.m_h1_b = in[18]; p.m_out_w = in[19]; p.m_out_b = in[20];

  p.Z = (float *)d_out;
  float *ws = (float *)d_ws;
  size_t off = 0;
  p.P      = ws + off; off += (size_t)NROWS * N_LAT;   // 263168
  p.V      = ws + off; off += (size_t)NROWS * N_G;     // 131584
  p.Kt     = ws + off; off += 2 * N_INN;
  p.g      = ws + off; off += N_G;
  p.h0     = ws + off; off += N_NEU;
  p.h1     = ws + off; off += N_NEU;
  p.logits = ws + off; off += NROWS + 7;
  p.a      = ws + off; off += NROWS + 7;
  p.bar    = (unsigned *)(ws + off);                   // 2 uints

  init_bar_kernel<<<1, 1, 0, stream>>>(p.bar);
  cell_kernel<<<NBLOCKS, TPB, 0, stream>>>(p);
}